// FeatureExtractor_63548336112350
// MI455X (gfx1250) — compile-verified
//
#include <hip/hip_runtime.h>

#define D 128
#define D_IN 7
#define NEG_SLOPE 0.2f
#define NLAYERS 5
#define MTILE 64          // nodes per block in the WMMA GEMM (4 x 16-row subtiles)

typedef __attribute__((ext_vector_type(2))) float v2f;
typedef __attribute__((ext_vector_type(4))) float v4f;
typedef __attribute__((ext_vector_type(8))) float v8f;

// ---------------- Layer 0 GEMM (K=7, too small for WMMA) ----------------
__global__ __launch_bounds__(256)
void gat_gemm0(const float* __restrict__ x,
               const float* __restrict__ Wl0, const float* __restrict__ Wr0,
               const float* __restrict__ bl0, const float* __restrict__ br0,
               float* __restrict__ xl, float* __restrict__ xr, int n)
{
    int idx = blockIdx.x * 256 + threadIdx.x;
    if (idx >= n * D) return;
    int nd = idx >> 7, d = idx & (D - 1);
    const float* W = blockIdx.y ? Wr0 : Wl0;
    const float* b = blockIdx.y ? br0 : bl0;
    float*       o = blockIdx.y ? xr  : xl;
    float s = b[d];
#pragma unroll
    for (int k = 0; k < D_IN; ++k)
        s += x[nd * D_IN + k] * W[k * D + d];
    o[idx] = s;
}

// ------------- Layers 1..4 GEMM: f32 WMMA 16x16x4, ReLU fused on A -------------
// Block = 256 threads = 8 waves; block computes MTILE(=64) nodes x full D=128.
// Wave w owns output columns [16w,16w+16) and 4 M-subtiles -> each B fragment
// is reused by 4 WMMAs (4x less L2 weight traffic than a 16-row tile).
// K=128 -> 32 k-steps -> 128 v_wmma_f32_16x16x4_f32 per wave.
__global__ __launch_bounds__(256)
void gat_gemm_wmma(const float* __restrict__ h,
                   const float* __restrict__ Wlp, const float* __restrict__ Wrp,
                   const float* __restrict__ blp, const float* __restrict__ brp,
                   float* __restrict__ xl, float* __restrict__ xr, int n)
{
    __shared__ float At[MTILE][132];              // 132 % 64 = 4 -> conflict-free
    const float* W    = (blockIdx.y == 0) ? Wlp : Wrp;
    const float* bias = (blockIdx.y == 0) ? blp : brp;
    float*       out  = (blockIdx.y == 0) ? xl  : xr;

    const int rowBase = blockIdx.x * MTILE;

    // stage MTILE x 128 A tile into LDS with inter-layer ReLU fused
    for (int i = threadIdx.x; i < MTILE * D; i += 256) {
        int m = i >> 7, k = i & (D - 1);
        int row = rowBase + m;
        row = row < n ? row : n - 1;              // clamp tail (stores are guarded)
        float v = h[row * D + k];
        At[m][k] = v > 0.f ? v : 0.f;
    }
    __syncthreads();

    const int wave  = threadIdx.x >> 5;
    const int lane  = threadIdx.x & 31;
    const int mrow  = lane & 15;                  // A: lanes 0-15 K0-1, lanes 16-31 K2-3
    const int khalf = (lane >> 4) * 2;            // 0 or 2
    const int col   = wave * 16 + (lane & 15);    // B/D column

    v8f acc[4];
#pragma unroll
    for (int s = 0; s < 4; ++s)
        acc[s] = (v8f){0.f, 0.f, 0.f, 0.f, 0.f, 0.f, 0.f, 0.f};

#pragma unroll
    for (int k = 0; k < D; k += 4) {
        v2f b;
        b.x = W[(k + khalf)     * D + col];
        b.y = W[(k + khalf + 1) * D + col];
#pragma unroll
        for (int s = 0; s < 4; ++s) {
            v2f a;
            a.x = At[s * 16 + mrow][k + khalf];
            a.y = At[s * 16 + mrow][k + khalf + 1];
            acc[s] = __builtin_amdgcn_wmma_f32_16x16x4_f32(false, a, false, b,
                                                           (short)0, acc[s],
                                                           false, false);
        }
    }

    const float bv  = bias[col];
    const int  rofs = (lane >> 4) << 3;           // lanes 16-31 hold rows M=8..15
#pragma unroll
    for (int s = 0; s < 4; ++s) {
#pragma unroll
        for (int r = 0; r < 8; ++r) {
            int row = rowBase + s * 16 + r + rofs;
            if (row < n)
                out[row * D + col] = acc[s][r] + bv;
        }
    }
}

// ---------------- per-layer state init ----------------
__global__ __launch_bounds__(256)
void gat_init(float* __restrict__ emax, float* __restrict__ denom,
              float* __restrict__ accum, int n)
{
    int idx = blockIdx.x * 256 + threadIdx.x;
    if (idx < n) { emax[idx] = -__builtin_inff(); denom[idx] = 0.f; }
    if (idx < n * D) accum[idx] = 0.f;
}

// ---------------- edge scores + segment max (wave per edge) ----------------
__global__ __launch_bounds__(256)
void gat_edge_score(const float* __restrict__ xl, const float* __restrict__ xr,
                    const int* __restrict__ ei, const float* __restrict__ att,
                    float* __restrict__ score, float* __restrict__ emax,
                    int eraw, int etot)
{
    int gid  = blockIdx.x * 256 + threadIdx.x;
    int e    = gid >> 5;
    int lane = gid & 31;
    if (e >= etot) return;
    int s_idx, d_idx;
    if (e < eraw) { s_idx = ei[e]; d_idx = ei[eraw + e]; }
    else          { s_idx = d_idx = e - eraw; }          // self loops

    v4f a = ((const v4f*)(xl + (size_t)s_idx * D))[lane];
    v4f b = ((const v4f*)(xr + (size_t)d_idx * D))[lane];
    v4f c = ((const v4f*)att)[lane];
    float s = 0.f;
#pragma unroll
    for (int i = 0; i < 4; ++i) {
        float v = a[i] + b[i];
        v = v > 0.f ? v : NEG_SLOPE * v;
        s += v * c[i];
    }
#pragma unroll
    for (int m = 16; m >= 1; m >>= 1) s += __shfl_xor(s, m, 32);
    if (lane == 0) {
        score[e] = s;
        atomicMax(emax + d_idx, s);                      // global_atomic_max_num_f32
    }
}

// ---------------- exp + segment sum (thread per edge) ----------------
__global__ __launch_bounds__(256)
void gat_edge_exp(const int* __restrict__ ei, float* __restrict__ score,
                  const float* __restrict__ emax, float* __restrict__ denom,
                  int eraw, int etot)
{
    int e = blockIdx.x * 256 + threadIdx.x;
    if (e >= etot) return;
    int d_idx = (e < eraw) ? ei[eraw + e] : (e - eraw);
    float ex = __expf(score[e] - emax[d_idx]);
    score[e] = ex;
    atomicAdd(denom + d_idx, ex);
}

// ---------------- weighted scatter-aggregate (wave per edge) ----------------
__global__ __launch_bounds__(256)
void gat_edge_agg(const float* __restrict__ xl, const int* __restrict__ ei,
                  const float* __restrict__ score, const float* __restrict__ denom,
                  float* __restrict__ accum, int eraw, int etot)
{
    int gid  = blockIdx.x * 256 + threadIdx.x;
    int e    = gid >> 5;
    int lane = gid & 31;
    if (e >= etot) return;
    int s_idx, d_idx;
    if (e < eraw) { s_idx = ei[e]; d_idx = ei[eraw + e]; }
    else          { s_idx = d_idx = e - eraw; }

    float alpha = score[e] / denom[d_idx];
    v4f v = ((const v4f*)(xl + (size_t)s_idx * D))[lane];
    float* dst = accum + (size_t)d_idx * D + lane * 4;   // L2-resident atomics
    atomicAdd(dst + 0, alpha * v[0]);
    atomicAdd(dst + 1, alpha * v[1]);
    atomicAdd(dst + 2, alpha * v[2]);
    atomicAdd(dst + 3, alpha * v[3]);
}

// ---------------- bias epilogue ----------------
__global__ __launch_bounds__(256)
void gat_finalize(const float* __restrict__ accum, const float* __restrict__ bias,
                  float* __restrict__ hout, int n)
{
    int idx = blockIdx.x * 256 + threadIdx.x;
    if (idx >= n * D) return;
    hout[idx] = accum[idx] + bias[idx & (D - 1)];
}

extern "C" void kernel_launch(void* const* d_in, const int* in_sizes, int n_in,
                              void* d_out, int out_size, void* d_ws, size_t ws_size,
                              hipStream_t stream)
{
    const float* x    = (const float*)d_in[0];
    const int*   ei   = (const int*)  d_in[1];
    const float* Wl0  = (const float*)d_in[2];
    const float* Wr0  = (const float*)d_in[3];
    const float* bl0  = (const float*)d_in[4];
    const float* br0  = (const float*)d_in[5];
    const float* Wl   = (const float*)d_in[6];
    const float* Wr   = (const float*)d_in[7];
    const float* bl   = (const float*)d_in[8];
    const float* br   = (const float*)d_in[9];
    const float* att  = (const float*)d_in[10];
    const float* bias = (const float*)d_in[11];

    const int N    = in_sizes[0] / D_IN;   // 50000
    const int E    = in_sizes[1] / 2;      // 800000
    const int ETOT = E + N;                // with self loops

    float* ws    = (float*)d_ws;
    float* h     = ws;
    float* xl    = h     + (size_t)N * D;
    float* xr    = xl    + (size_t)N * D;
    float* acc   = xr    + (size_t)N * D;
    float* scr   = acc   + (size_t)N * D;
    float* emax  = scr   + ETOT;
    float* denom = emax  + N;

    dim3 blk(256);
    const int ndBlocks     = (N * D + 255) / 256;
    const int gemmBlocks   = (N + MTILE - 1) / MTILE;
    const int edgeWaveBlks = (ETOT * 32 + 255) / 256;
    const int edgeBlks     = (ETOT + 255) / 256;

    for (int l = 0; l < NLAYERS; ++l) {
        if (l == 0) {
            gat_gemm0<<<dim3(ndBlocks, 2), blk, 0, stream>>>(
                x, Wl0, Wr0, bl0, br0, xl, xr, N);
        } else {
            gat_gemm_wmma<<<dim3(gemmBlocks, 2), blk, 0, stream>>>(
                h,
                Wl + (size_t)(l - 1) * D * D, Wr + (size_t)(l - 1) * D * D,
                bl + (size_t)(l - 1) * D,     br + (size_t)(l - 1) * D,
                xl, xr, N);
        }
        gat_init<<<ndBlocks, blk, 0, stream>>>(emax, denom, acc, N);
        gat_edge_score<<<edgeWaveBlks, blk, 0, stream>>>(
            xl, xr, ei, att + (size_t)l * D, scr, emax, E, ETOT);
        gat_edge_exp<<<edgeBlks, blk, 0, stream>>>(ei, scr, emax, denom, E, ETOT);
        gat_edge_agg<<<edgeWaveBlks, blk, 0, stream>>>(xl, ei, scr, denom, acc, E, ETOT);
        float* hout = (l == NLAYERS - 1) ? (float*)d_out : h;
        gat_finalize<<<ndBlocks, blk, 0, stream>>>(acc, bias + (size_t)l * D, hout, N);
    }
}